// FFM_layer_17858474017258
// MI455X (gfx1250) — compile-verified
//
#include <hip/hip_runtime.h>
#include <math.h>

// FFM layer for MI455X (gfx1250, wave32).
// Memory-bound: ~133 MB of random 1248-B embedding-row gathers per call
// (~6 us at 23.3 TB/s HBM). Dense GEMM (4096x13 @ 13x312, ~33 MFLOP) is done
// with V_WMMA_F32_16X16X4_F32 per 16-row batch tile, staged through LDS, and
// its load/WMMA latency is hidden under the gather loop (barrier comes after
// the bulk gather traffic has been issued). Gather indices are wave-uniform,
// so they are scalarized with v_readlane into SGPR bases: SALU address math,
// loop-invariant lane offsets, no ds_bpermute in the address chain.

typedef float v2f __attribute__((ext_vector_type(2)));
typedef float v8f __attribute__((ext_vector_type(8)));

#define N_DENSE   13
#define N_SPARSE  26
#define VOCAB     10000
#define N_FIELD   39
#define KDIM      8
#define ROWF      (N_FIELD * KDIM)   // 312 floats per v-row
#define ROWF4     (ROWF / 4)         // 78 float4 per v-row (exact)
#define PADF      320                // padded to 20 WMMA column tiles of 16
#define PADF4     (PADF / 4)         // 80
#define NTILES    (PADF / 16)        // 20
#define WAVES     16                 // 16 waves -> 16 batch rows per block
#define BLOCK     (WAVES * 32)       // 512 threads

__global__ __launch_bounds__(BLOCK)
void ffm_fused_kernel(const float* __restrict__ dense_x,   // (B,13)
                      const int*   __restrict__ sparse_x,  // (B,26)
                      const float* __restrict__ w0,        // (1,)
                      const float* __restrict__ w,         // (260013,1)
                      const float* __restrict__ v,         // (260013,39,8)
                      float* __restrict__ out,             // (B,1)
                      int batch)
{
    // dense contribution field_f for this block's 16 rows, padded to 320 cols
    __shared__ __align__(16) float dacc[WAVES][PADF];   // 20 KB

    const int tid  = threadIdx.x;
    const int lane = tid & 31;
    const int wave = tid >> 5;
    const int row0 = blockIdx.x * WAVES;

    const int b  = row0 + wave;
    const int bc = (b < batch) ? b : (batch - 1);       // clamped (loads only)

    // issue the per-row sparse ids + linear-term gathers early (latency hides
    // under everything that follows)
    int sxv = (lane < N_SPARSE) ? sparse_x[bc * N_SPARSE + lane] : 0;
    float lp = 0.0f;
    if (lane < N_DENSE)  lp += dense_x[bc * N_DENSE + lane] * w[lane];
    if (lane < N_SPARSE) lp += w[N_DENSE + lane * VOCAB + sxv];

    // ---------------- Phase 1: dense GEMM via V_WMMA_F32_16X16X4_F32 --------
    // A (16x4 f32):  lane l -> m = l&15, koff = (l>>4)*2 ; a = {A[m][k0], A[m][k0+1]}
    // B (4x16 f32):  lane l -> n = l&15, koff = (l>>4)*2 ; b = {B[k0][n], B[k0+1][n]}
    // C/D (16x16):   c[r] = D[r + (l>>4)*8][l&15]
    // All out-of-range loads are branchless: clamped-address load + cndmask 0.
    {
        const int mn    = lane & 15;
        const int koff  = (lane >> 4) << 1;
        const int rbase = (lane >> 4) << 3;

        // A fragments are identical for every column tile: hoist (8 loads)
        v2f afrag[4];
#pragma unroll
        for (int i = 0; i < 4; ++i) {
            const int k0 = 4 * i + koff;
            const int k1 = k0 + 1;
            const float ax = dense_x[(row0 + mn) * N_DENSE + (k0 < N_DENSE ? k0 : 0)];
            const float ay = dense_x[(row0 + mn) * N_DENSE + (k1 < N_DENSE ? k1 : 0)];
            afrag[i].x = (k0 < N_DENSE) ? ax : 0.0f;
            afrag[i].y = (k1 < N_DENSE) ? ay : 0.0f;
        }

        for (int t = wave; t < NTILES; t += WAVES) {    // 20 tiles over 16 waves
            const int col0 = t << 4;
            const int cn   = col0 + mn;
            const int cnc  = (cn < ROWF) ? cn : 0;
            v2f bfrag[4];
#pragma unroll
            for (int i = 0; i < 4; ++i) {               // 8 batched loads
                const int k0 = 4 * i + koff;
                const int k1 = k0 + 1;
                const float bx = v[(k0 < N_DENSE ? k0 : 0) * ROWF + cnc];
                const float by = v[(k1 < N_DENSE ? k1 : 0) * ROWF + cnc];
                bfrag[i].x = (k0 < N_DENSE && cn < ROWF) ? bx : 0.0f;
                bfrag[i].y = (k1 < N_DENSE && cn < ROWF) ? by : 0.0f;
            }
            v8f c = {};
#pragma unroll
            for (int i = 0; i < 4; ++i)                 // back-to-back WMMAs
                c = __builtin_amdgcn_wmma_f32_16x16x4_f32(
                        /*neg_a=*/false, afrag[i], /*neg_b=*/false, bfrag[i],
                        /*c_mod=*/(short)0, c, /*reuse_a=*/false, /*reuse_b=*/false);
#pragma unroll
            for (int r = 0; r < 8; ++r)
                dacc[rbase + r][cn] = c[r];             // cols >=312 exact zeros
        }
    }

    // ---------------- Phase 2: wave-per-row embedding gather ----------------
    // accumulators: element e = 4*(lane + 32*j) + c  => k = 4*(lane&1) + c
    float4 a0, a1, a2;
    a0.x = a0.y = a0.z = a0.w = 0.0f;
    a1.x = a1.y = a1.z = a1.w = 0.0f;
    a2.x = a2.y = a2.z = a2.w = 0.0f;

    const bool full3 = (lane + 64) < ROWF4;               // lane < 14
    const int  q2    = full3 ? (lane + 64) : (ROWF4 - 1); // clamped in-row index

    // fully unrolled: readlane(s) -> SGPR index, scalar base math, three b128
    // loads per feature with loop-invariant lane offsets
#pragma unroll
    for (int s = 0; s < N_SPARSE; ++s) {
        const int sx = __builtin_amdgcn_readlane(sxv, s);           // uniform
        const float4* rp =
            (const float4*)(v + (size_t)(N_DENSE + s * VOCAB + sx) * ROWF);
        float4 r0 = rp[lane];                           // coalesced b128 loads:
        float4 r1 = rp[lane + 32];                      // 1248 contiguous B/wave
        float4 r2 = rp[q2];
        // branchless tail mask (cndmask, no exec divergence)
        r2.x = full3 ? r2.x : 0.0f;
        r2.y = full3 ? r2.y : 0.0f;
        r2.z = full3 ? r2.z : 0.0f;
        r2.w = full3 ? r2.w : 0.0f;
        a0.x += r0.x; a0.y += r0.y; a0.z += r0.z; a0.w += r0.w;
        a1.x += r1.x; a1.y += r1.y; a1.z += r1.z; a1.w += r1.w;
        a2.x += r2.x; a2.y += r2.y; a2.z += r2.z; a2.w += r2.w;
    }

    // dense contribution becomes visible here; bulk traffic already issued
    __syncthreads();
    {
        const float4* drow = (const float4*)(&dacc[wave][0]);
        float4 d0 = drow[lane];
        float4 d1 = drow[lane + 32];
        float4 d2 = drow[(lane + 64 < PADF4) ? (lane + 64) : (PADF4 - 1)];
        const bool dok = (lane + 64) < PADF4;           // lanes 0..15
        d2.x = dok ? d2.x : 0.0f;
        d2.y = dok ? d2.y : 0.0f;
        d2.z = dok ? d2.z : 0.0f;
        d2.w = dok ? d2.w : 0.0f;
        a0.x += d0.x; a0.y += d0.y; a0.z += d0.z; a0.w += d0.w;
        a1.x += d1.x; a1.y += d1.y; a1.z += d1.z; a1.w += d1.w;
        a2.x += d2.x; a2.y += d2.y; a2.z += d2.z; a2.w += d2.w;
    }

    // ---------------- Phase 3: reductions ----------------
    // per-component partial s_k (k fixed per component per lane parity)
    float skx = a0.x + a1.x + a2.x;
    float sky = a0.y + a1.y + a2.y;
    float skz = a0.z + a1.z + a2.z;
    float skw = a0.w + a1.w + a2.w;

    // sum of squares of all field_f elements (padding entries are exact zeros)
    float ss = a0.x*a0.x + a0.y*a0.y + a0.z*a0.z + a0.w*a0.w
             + a1.x*a1.x + a1.y*a1.y + a1.z*a1.z + a1.w*a1.w
             + a2.x*a2.x + a2.y*a2.y + a2.z*a2.z + a2.w*a2.w;

    // reduce s_k within each lane-parity class (masks preserve bit0)
#pragma unroll
    for (int m = 2; m <= 16; m <<= 1) {
        skx += __shfl_xor(skx, m, 32);
        sky += __shfl_xor(sky, m, 32);
        skz += __shfl_xor(skz, m, 32);
        skw += __shfl_xor(skw, m, 32);
    }
    float q = skx*skx + sky*sky + skz*skz + skw*skw;    // sum s_k^2 over 4 k's
    q += __shfl_xor(q, 1, 32);                          // + other parity's 4 k's

#pragma unroll
    for (int m = 1; m <= 16; m <<= 1) ss += __shfl_xor(ss, m, 32);

    // linear term partials (issued at kernel start) -> full wave reduce
#pragma unroll
    for (int m = 1; m <= 16; m <<= 1) lp += __shfl_xor(lp, m, 32);

    if (lane == 0 && b < batch) {
        const float z = lp + w0[0] + 0.5f * (q - ss);
        out[b] = 1.0f / (1.0f + expf(-z));
    }
}

extern "C" void kernel_launch(void* const* d_in, const int* in_sizes, int n_in,
                              void* d_out, int out_size, void* d_ws, size_t ws_size,
                              hipStream_t stream) {
    const float* dense_x  = (const float*)d_in[0];
    const int*   sparse_x = (const int*)d_in[1];
    const float* w0       = (const float*)d_in[2];
    const float* w        = (const float*)d_in[3];
    const float* v        = (const float*)d_in[4];
    float*       out      = (float*)d_out;

    const int batch  = in_sizes[0] / N_DENSE;           // 4096
    const int blocks = (batch + WAVES - 1) / WAVES;     // 256

    hipLaunchKernelGGL(ffm_fused_kernel, dim3(blocks), dim3(BLOCK), 0, stream,
                       dense_x, sparse_x, w0, w, v, out, batch);
}